// GIN_6674379178026
// MI455X (gfx1250) — compile-verified
//
#include <hip/hip_runtime.h>

#define N_NODES 50000
#define N_EDGES 800000
#define DIM     200
#define NCLS    20

typedef __attribute__((ext_vector_type(2))) float v2f;
typedef __attribute__((ext_vector_type(8))) float v8f;

__global__ void zero_f32(float* __restrict__ p, int n) {
  int i = blockIdx.x * blockDim.x + threadIdx.x;
  if (i < n) p[i] = 0.0f;
}

// One thread per (edge, d): out[rows[e], d] += vals[e] * dense[cols[e], d]
__global__ void spmm_scatter(const float* __restrict__ vals,
                             const int*   __restrict__ rows,
                             const int*   __restrict__ cols,
                             const float* __restrict__ dense,
                             float*       __restrict__ out,
                             int D, int total) {
  int i = blockIdx.x * blockDim.x + threadIdx.x;
  if (i >= total) return;
  int e = i / D;
  int d = i - e * D;
  int r = rows[e];
  int c = cols[e];
  float v = vals[e];
  atomicAdd(&out[(size_t)r * D + d], v * dense[(size_t)c * D + d]);
}

__global__ void relu_bias_inplace(float* __restrict__ h,
                                  const float* __restrict__ b,
                                  int n, int D) {
  int i = blockIdx.x * blockDim.x + threadIdx.x;
  if (i >= n) return;
  float x = h[i] + b[i % D];
  h[i] = x > 0.0f ? x : 0.0f;
}

// g[N,20] = h[N,200] @ W1[200,20], fp32 WMMA 16x16x4.
// One wave -> 16 rows x 32 cols (cols 20..31 are zero padding in LDS).
__global__ void __launch_bounds__(256)
gemm_hW1_wmma(const float* __restrict__ h,
              const float* __restrict__ W1,
              float* __restrict__ g, int nTiles) {
  __shared__ float w[DIM][32];           // W1 zero-padded to 32 cols
  for (int i = threadIdx.x; i < DIM * 32; i += 256) {
    int r = i >> 5, c = i & 31;
    w[r][c] = (c < NCLS) ? W1[r * NCLS + c] : 0.0f;
  }
  __syncthreads();

  int wave = (int)((blockIdx.x * 256u + threadIdx.x) >> 5);
  if (wave >= nTiles) return;            // wave-uniform: EXEC stays all-1s below

  int lane = threadIdx.x & 31;
  int half = lane >> 4;                  // 0: lanes 0-15 (K=0,1), 1: lanes 16-31 (K=2,3)
  int l    = lane & 15;
  const float* hrow = h + (size_t)(wave * 16 + l) * DIM;  // A: M = l

  v8f acc0 = {};  // cols 0..15
  v8f acc1 = {};  // cols 16..31 (only 16..19 valid)
  for (int kb = 0; kb < DIM; kb += 4) {
    int k0 = kb + 2 * half;
    v2f a, b0, b1;
    a.x  = hrow[k0];
    a.y  = hrow[k0 + 1];
    b0.x = w[k0][l];          b0.y = w[k0 + 1][l];
    b1.x = w[k0][16 + l];     b1.y = w[k0 + 1][16 + l];
    acc0 = __builtin_amdgcn_wmma_f32_16x16x4_f32(false, a, false, b0,
                                                 (short)0, acc0, false, false);
    acc1 = __builtin_amdgcn_wmma_f32_16x16x4_f32(false, a, false, b1,
                                                 (short)0, acc1, false, false);
  }

  size_t row0 = (size_t)wave * 16;
#pragma unroll
  for (int r = 0; r < 8; ++r) {
    size_t row = row0 + (size_t)(r + 8 * half);  // C/D layout: VGPR r -> M = r + 8*half
    g[row * NCLS + l] = acc0[r];                 // N = l (0..15)
    int c1 = 16 + l;
    if (c1 < NCLS) g[row * NCLS + c1] = acc1[r]; // N = 16..19
  }
}

__global__ void bias_add(float* __restrict__ out, const float* __restrict__ b,
                         int n, int C) {
  int i = blockIdx.x * blockDim.x + threadIdx.x;
  if (i >= n) return;
  out[i] += b[i % C];
}

extern "C" void kernel_launch(void* const* d_in, const int* in_sizes, int n_in,
                              void* d_out, int out_size, void* d_ws, size_t ws_size,
                              hipStream_t stream) {
  // setup_inputs order: x, vals0, vals1, W0, b0, W1, b1, rows0, cols0, rows1, cols1
  const float* vals0 = (const float*)d_in[1];
  const float* vals1 = (const float*)d_in[2];
  const float* W0    = (const float*)d_in[3];
  const float* b0    = (const float*)d_in[4];
  const float* W1    = (const float*)d_in[5];
  const float* b1    = (const float*)d_in[6];
  const int*   rows0 = (const int*)d_in[7];
  const int*   cols0 = (const int*)d_in[8];
  const int*   rows1 = (const int*)d_in[9];
  const int*   cols1 = (const int*)d_in[10];
  float* out = (float*)d_out;

  // Workspace: ax0/h [N,DIM] then g [N,NCLS]  (44 MB total)
  float* ax0 = (float*)d_ws;
  float* g   = ax0 + (size_t)N_NODES * DIM;

  const int nAX  = N_NODES * DIM;   // 10,000,000
  const int nOut = N_NODES * NCLS;  //  1,000,000

  zero_f32<<<(nAX  + 255) / 256, 256, 0, stream>>>(ax0, nAX);
  zero_f32<<<(nOut + 255) / 256, 256, 0, stream>>>(out, nOut);

  // Layer 0: ax0 = A0 @ W0 + A1 @ W0   (gather W0 rows, scatter-add)
  const int tot200 = N_EDGES * DIM;  // 160,000,000
  spmm_scatter<<<(tot200 + 255) / 256, 256, 0, stream>>>(vals0, rows0, cols0,
                                                         W0, ax0, DIM, tot200);
  spmm_scatter<<<(tot200 + 255) / 256, 256, 0, stream>>>(vals1, rows1, cols1,
                                                         W0, ax0, DIM, tot200);

  // h = relu(ax0 + b0), in place
  relu_bias_inplace<<<(nAX + 255) / 256, 256, 0, stream>>>(ax0, b0, nAX, DIM);

  // Algebraic rewrite: (A h) W1 == A (h W1).  Dense WMMA GEMM first:
  const int nTiles = N_NODES / 16;   // 3125 (exact)
  gemm_hW1_wmma<<<(nTiles + 7) / 8, 256, 0, stream>>>(ax0, W1, g, nTiles);

  // Layer 1 scatter on the narrow (20-wide) g -> 10x less atomic traffic
  const int tot20 = N_EDGES * NCLS;  // 16,000,000
  spmm_scatter<<<(tot20 + 255) / 256, 256, 0, stream>>>(vals0, rows0, cols0,
                                                        g, out, NCLS, tot20);
  spmm_scatter<<<(tot20 + 255) / 256, 256, 0, stream>>>(vals1, rows1, cols1,
                                                        g, out, NCLS, tot20);

  bias_add<<<(nOut + 255) / 256, 256, 0, stream>>>(out, b1, nOut, NCLS);
}